// TemporalGNN_90314572301066
// MI455X (gfx1250) — compile-verified
//
#include <hip/hip_runtime.h>
#include <hip/hip_bf16.h>

typedef __attribute__((ext_vector_type(2))) float v2f;
typedef __attribute__((ext_vector_type(8))) float v8f;

#define BN_EPS 1e-5f

// ---------------- utility kernels ----------------

__global__ void gnn_zero_f32(float* __restrict__ p, long n) {
    long t = (long)blockIdx.x * blockDim.x + threadIdx.x;
    if (t < n) p[t] = 0.0f;
}

// degree accumulation over real edges (self-loop contribution added in finalize)
__global__ void gnn_edge_deg(const int* __restrict__ dst, float* __restrict__ deg, int E) {
    int e = blockIdx.x * blockDim.x + threadIdx.x;
    if (e < E) atomicAdd(&deg[dst[e]], 1.0f);
}

// dis[i] = rsqrt(deg_edges[i] + 1)   (+1 = self loop; always > 0)
__global__ void gnn_finalize_dis(float* __restrict__ deg, int n) {
    int i = blockIdx.x * blockDim.x + threadIdx.x;
    if (i < n) deg[i] = rsqrtf(deg[i] + 1.0f);
}

// per-column sum / sum-of-squares; thread == column (coalesced row reads)
__global__ void gnn_colstats(const float* __restrict__ X, float* __restrict__ sum,
                             float* __restrict__ sumsq, int nrows, int ncols) {
    int c = threadIdx.x;
    if (c >= ncols) return;
    float s = 0.0f, q = 0.0f;
    for (int r = blockIdx.x; r < nrows; r += gridDim.x) {
        float v = X[(long)r * ncols + c];
        s += v;
        q += v * v;
    }
    atomicAdd(&sum[c], s);
    atomicAdd(&sumsq[c], q);
}

// scale = g * rsqrt(var+eps), shift = be - mean*scale   (biased var, matches jnp.var)
__global__ void gnn_finalize_affine(const float* __restrict__ sum, const float* __restrict__ sumsq,
                                    const float* __restrict__ g, const float* __restrict__ be,
                                    float* __restrict__ scale, float* __restrict__ shift,
                                    int nrows, int ncols) {
    int c = threadIdx.x;
    if (c >= ncols) return;
    float inv = 1.0f / (float)nrows;
    float m = sum[c] * inv;
    float v = sumsq[c] * inv - m * m;
    float sc = g[c] * rsqrtf(v + BN_EPS);
    scale[c] = sc;
    shift[c] = be[c] - m * sc;
}

// ---------------- fused BN(+ReLU) -> WMMA f32 GEMM (LDS-staged A tile) ----------------
// D[nrows x OUT] = f(A) @ W, where f(A[r][k]) = relu?(A[r][k]*scale[k]+shift[k]).
// Block = 8 waves sharing one 16-row A tile (BN applied once at LDS fill, coalesced),
// each wave owns one 16-col n-tile; V_WMMA_F32_16X16X4_F32 over K.
// LDS row stride K+2: 8B-aligned ds_load_b64 fragments, bank-conflict-free.
__global__ void gnn_gemm_bn_wmma(const float* __restrict__ A, const float* __restrict__ W,
                                 const float* __restrict__ scale, const float* __restrict__ shift,
                                 float* __restrict__ D, int nrows, int K, int kshift,
                                 int OUT, int relu_a) {
    __shared__ float As[16 * 258];          // up to K=256, stride K+2
    const int LDA = K + 2;
    int m0 = blockIdx.y * 16;

    // ---- coalesced fill: 16 rows x K cols, BN(+ReLU) applied here ----
    int total = 16 << kshift;               // 16*K
    int kmask = K - 1;
    for (int t = threadIdx.x; t < total; t += blockDim.x) {
        int r = t >> kshift;
        int k = t & kmask;
        int row = m0 + r;
        if (row >= nrows) row = nrows - 1;  // clamp (stores are guarded later)
        float v = A[(long)row * K + k] * scale[k] + shift[k];
        if (relu_a) v = fmaxf(v, 0.0f);
        As[r * LDA + k] = v;
    }
    __syncthreads();

    int lane = threadIdx.x & 31;
    int wave = threadIdx.x >> 5;
    int n0 = (blockIdx.x * (blockDim.x >> 5) + wave) * 16;   // wave-uniform
    if (n0 >= OUT) return;                  // uniform exit: EXEC stays all-ones
    int half = lane >> 4;                   // 0: K pair {0,1}; 1: K pair {2,3}
    int idx  = lane & 15;
    const float* Alane = As + idx * LDA;

    v8f acc = {};
    for (int k0 = 0; k0 < K; k0 += 4) {
        int kb = k0 + 2 * half;
        v2f a = {Alane[kb], Alane[kb + 1]};
        v2f b = {W[(long)kb * OUT + n0 + idx],
                 W[(long)(kb + 1) * OUT + n0 + idx]};
        acc = __builtin_amdgcn_wmma_f32_16x16x4_f32(
            /*neg_a=*/false, a, /*neg_b=*/false, b,
            /*c_mod=*/(short)0, acc, /*reuse_a=*/false, /*reuse_b=*/false);
    }
    // C/D layout: VGPR i -> M = i (lanes 0-15) / 8+i (lanes 16-31), N = lane&15
    for (int i = 0; i < 8; ++i) {
        int m = m0 + i + 8 * half;
        if (m < nrows) D[(long)m * OUT + n0 + idx] = acc[i];
    }
}

// ---------------- aggregation ----------------

// agg[i][f] = H[i][f] * dis[i]^2   (self-loop term; also initializes agg)
__global__ void gnn_selfloop_init(const float* __restrict__ H, const float* __restrict__ dis,
                                  float* __restrict__ agg, long total, int F) {
    long t = (long)blockIdx.x * blockDim.x + threadIdx.x;
    if (t < total) {
        long r = t / F;
        float d = dis[r];
        agg[t] = H[t] * d * d;
    }
}

// one thread per (edge, float4 chunk): coalesced gather + 4 f32 global atomics
__global__ void gnn_edge_agg(const float* __restrict__ H, const int* __restrict__ src,
                             const int* __restrict__ dst, const float* __restrict__ dis,
                             float* __restrict__ agg, int E, int F) {
    int f4cnt = F >> 2;
    long t = (long)blockIdx.x * blockDim.x + threadIdx.x;
    long total = (long)E * f4cnt;
    if (t >= total) return;
    int e  = (int)(t / f4cnt);
    int f4 = (int)(t % f4cnt);
    int s = src[e], d = dst[e];
    float w = dis[s] * dis[d];
    float4 v = ((const float4*)(H + (long)s * F))[f4];
    float* ap = agg + (long)d * F + (long)f4 * 4;
    atomicAdd(ap + 0, v.x * w);
    atomicAdd(ap + 1, v.y * w);
    atomicAdd(ap + 2, v.z * w);
    atomicAdd(ap + 3, v.w * w);
}

// out = relu(X*scale + shift)
__global__ void gnn_bn_relu_out(const float* __restrict__ X, const float* __restrict__ scale,
                                const float* __restrict__ shift, float* __restrict__ out,
                                long total, int F) {
    long t = (long)blockIdx.x * blockDim.x + threadIdx.x;
    if (t < total) {
        int c = (int)(t % F);
        out[t] = fmaxf(X[t] * scale[c] + shift[c], 0.0f);
    }
}

// ---------------- host orchestration ----------------

static inline int cdiv_l(long a, long b) { return (int)((a + b - 1) / b); }
static inline int ilog2_i(int v) { int s = 0; while ((1 << s) < v) ++s; return s; }

extern "C" void kernel_launch(void* const* d_in, const int* in_sizes, int n_in,
                              void* d_out, int out_size, void* d_ws, size_t ws_size,
                              hipStream_t stream) {
    const float* x    = (const float*)d_in[0];
    const int*   ei   = (const int*)d_in[1];
    const float* g_in = (const float*)d_in[2];
    const float* be_in= (const float*)d_in[3];
    const float* w1   = (const float*)d_in[4];
    // d_in[5] = b1: constant per-feature bias cancels inside BatchNorm -> unused
    const float* g1   = (const float*)d_in[6];
    const float* be1  = (const float*)d_in[7];
    const float* w2   = (const float*)d_in[8];
    // d_in[9] = b2: cancels inside BatchNorm -> unused
    const float* g2   = (const float*)d_in[10];
    const float* be2  = (const float*)d_in[11];

    int F  = in_sizes[2];            // 128
    int n  = in_sizes[0] / F;        // 50000
    int E  = in_sizes[1] / 2;        // 800000
    int H2 = in_sizes[4] / F;        // 256
    int H  = in_sizes[10];           // 128

    const int* src = ei;
    const int* dst = ei + E;

    float* ws    = (float*)d_ws;
    float* dis   = ws;                         // n
    float* sum   = ws + n;                     // 256
    float* sumsq = sum + 256;                  // 256
    float* scale = sumsq + 256;                // 256
    float* shift = scale + 256;                // 256
    float* h1    = shift + 256;                // n*H2
    float* agg1  = h1 + (long)n * H2;          // n*H2
    float* h2    = h1;                         // n*H   (aliases dead h1)
    float* agg2  = h1 + (long)n * H;           // n*H   (aliases dead h1)

    const int B = 256;
    int mtiles = cdiv_l(n, 16);                // one 16-row tile per block (y)

    // degrees -> deg_inv_sqrt
    gnn_zero_f32<<<cdiv_l(n, B), B, 0, stream>>>(dis, n);
    gnn_edge_deg<<<cdiv_l(E, B), B, 0, stream>>>(dst, dis, E);
    gnn_finalize_dis<<<cdiv_l(n, B), B, 0, stream>>>(dis, n);

    // BN0 over x (fused into GEMM1 LDS fill)
    gnn_zero_f32<<<cdiv_l(512, B), B, 0, stream>>>(sum, 512);
    gnn_colstats<<<512, F, 0, stream>>>(x, sum, sumsq, n, F);
    gnn_finalize_affine<<<1, F, 0, stream>>>(sum, sumsq, g_in, be_in, scale, shift, n, F);

    // layer 1 GEMM: h1 = BN0(x) @ w1   (block covers 16 rows x 128 cols)
    gnn_gemm_bn_wmma<<<dim3(cdiv_l(H2, 128), mtiles), 256, 0, stream>>>(
        x, w1, scale, shift, h1, n, F, ilog2_i(F), H2, 0);

    // layer 1 aggregation: agg1 = D^-1/2 (A+I) D^-1/2 h1
    gnn_selfloop_init<<<cdiv_l((long)n * H2, B), B, 0, stream>>>(h1, dis, agg1, (long)n * H2, H2);
    gnn_edge_agg<<<cdiv_l((long)E * (H2 / 4), B), B, 0, stream>>>(h1, src, dst, dis, agg1, E, H2);

    // BN1 (+ReLU) fused into GEMM2 LDS fill
    gnn_zero_f32<<<cdiv_l(512, B), B, 0, stream>>>(sum, 512);
    gnn_colstats<<<512, H2, 0, stream>>>(agg1, sum, sumsq, n, H2);
    gnn_finalize_affine<<<1, H2, 0, stream>>>(sum, sumsq, g1, be1, scale, shift, n, H2);

    // layer 2 GEMM: h2 = relu(BN1(agg1)) @ w2
    gnn_gemm_bn_wmma<<<dim3(cdiv_l(H, 128), mtiles), 256, 0, stream>>>(
        agg1, w2, scale, shift, h2, n, H2, ilog2_i(H2), H, 1);

    // layer 2 aggregation
    gnn_selfloop_init<<<cdiv_l((long)n * H, B), B, 0, stream>>>(h2, dis, agg2, (long)n * H, H);
    gnn_edge_agg<<<cdiv_l((long)E * (H / 4), B), B, 0, stream>>>(h2, src, dst, dis, agg2, E, H);

    // BN2 + ReLU -> output
    gnn_zero_f32<<<cdiv_l(512, B), B, 0, stream>>>(sum, 512);
    gnn_colstats<<<512, H, 0, stream>>>(agg2, sum, sumsq, n, H);
    gnn_finalize_affine<<<1, H, 0, stream>>>(sum, sumsq, g2, be2, scale, shift, n, H);
    gnn_bn_relu_out<<<cdiv_l((long)n * H, B), B, 0, stream>>>(agg2, scale, shift, (float*)d_out, (long)n * H, H);
}